// GraphSAGE_87582973100260
// MI455X (gfx1250) — compile-verified
//
#include <hip/hip_runtime.h>
#include <hip/hip_bf16.h>

typedef float v2f __attribute__((ext_vector_type(2)));
typedef float v8f __attribute__((ext_vector_type(8)));

#define D_FEAT 64

// deg[i] = 1 (self loop)
__global__ void sg_init_deg(float* __restrict__ deg, int n) {
    int i = blockIdx.x * blockDim.x + threadIdx.x;
    if (i < n) deg[i] = 1.0f;
}

__global__ void sg_accum_deg(const int* __restrict__ col, float* __restrict__ deg, int e) {
    int i = blockIdx.x * blockDim.x + threadIdx.x;
    if (i < e) atomicAdd(&deg[col[i]], 1.0f);
}

__global__ void sg_rsqrt(float* __restrict__ deg, int n) {
    int i = blockIdx.x * blockDim.x + threadIdx.x;
    if (i < n) deg[i] = rsqrtf(deg[i]);
}

// Initialize h_out with the self-loop term: h_out[i,:] = dinv[i]^2 * h_in[i,:]
__global__ void sg_self(const float* __restrict__ hin, const float* __restrict__ dinv,
                        float* __restrict__ hout, int n) {
    long long idx = (long long)blockIdx.x * blockDim.x + threadIdx.x;
    if (idx >= (long long)n * D_FEAT) return;
    int i = (int)(idx >> 6);
    float d = dinv[i];
    hout[idx] = d * d * hin[idx];
}

// 64 consecutive threads handle one edge (one feature each): coalesced gather
// of h_in[row,:] and contiguous f32 atomic scatter-add into h_out[col,:].
__global__ void sg_edges(const int* __restrict__ row, const int* __restrict__ col,
                         const float* __restrict__ dinv, const float* __restrict__ hin,
                         float* __restrict__ hout, int e) {
    long long idx = (long long)blockIdx.x * blockDim.x + threadIdx.x;
    if (idx >= (long long)e * D_FEAT) return;
    int ei = (int)(idx >> 6);
    int f  = (int)(idx & 63);
    int r = row[ei];
    int c = col[ei];
    float w = dinv[r] * dinv[c];
    atomicAdd(&hout[(long long)c * D_FEAT + f], w * hin[(long long)r * D_FEAT + f]);
}

// out = relu(h @ W^T + b) via V_WMMA_F32_16X16X4_F32.
// Block = 128 threads = 4 waves; wave w owns N-tile w (n0 = 16*w) of M-tile blockIdx.x.
// A-layout (16x4 f32): lane L holds M = L%16, VGPRs = K = 2*(L/16), 2*(L/16)+1.
// B-layout (4x16 f32): lane L holds N = L%16, same K striping -> since
// B[k][n] = W[n][k], B loads use identical addressing on W's rows.
// C/D (16x16 f32): lane L, VGPR v -> M = v + 8*(L/16), N = L%16.
__global__ void sg_gemm_relu_wmma(const float* __restrict__ h, const float* __restrict__ W,
                                  const float* __restrict__ bias, float* __restrict__ out) {
    const int lane = threadIdx.x & 31;
    const int wave = threadIdx.x >> 5;
    const int m0   = blockIdx.x * 16;
    const int n0   = wave * 16;
    const int half = lane >> 4;
    const int l16  = lane & 15;

    const float* arow = h + (long long)(m0 + l16) * D_FEAT;  // A row (k contiguous)
    const float* brow = W + (n0 + l16) * D_FEAT;             // W row n (k contiguous)

    v8f c = {0.f, 0.f, 0.f, 0.f, 0.f, 0.f, 0.f, 0.f};
#pragma unroll
    for (int k = 0; k < D_FEAT; k += 4) {
        const int kk = k + 2 * half;
        v2f a, b;
        a.x = arow[kk];     a.y = arow[kk + 1];
        b.x = brow[kk];     b.y = brow[kk + 1];
        c = __builtin_amdgcn_wmma_f32_16x16x4_f32(false, a, false, b, (short)0, c,
                                                  false, false);
    }

    const float bn = bias[n0 + l16];
#pragma unroll
    for (int v = 0; v < 8; ++v) {
        const int m = m0 + v + 8 * half;
        float r = c[v] + bn;
        out[(long long)m * D_FEAT + n0 + l16] = r > 0.0f ? r : 0.0f;
    }
}

// Scalar tail for M not divisible by 16 (not hit for N=100000, kept for safety).
__global__ void sg_gemm_relu_tail(const float* __restrict__ h, const float* __restrict__ W,
                                  const float* __restrict__ bias, float* __restrict__ out,
                                  int mstart, int n) {
    int m = mstart + blockIdx.x;
    if (m >= n) return;
    int f = threadIdx.x;  // 0..63
    float acc = bias[f];
    const float* hr = h + (long long)m * D_FEAT;
    const float* wr = W + f * D_FEAT;
    for (int k = 0; k < D_FEAT; ++k) acc += hr[k] * wr[k];
    out[(long long)m * D_FEAT + f] = acc > 0.0f ? acc : 0.0f;
}

extern "C" void kernel_launch(void* const* d_in, const int* in_sizes, int n_in,
                              void* d_out, int out_size, void* d_ws, size_t ws_size,
                              hipStream_t stream) {
    const float* x    = (const float*)d_in[0];   // [N, 64]
    const float* W    = (const float*)d_in[1];   // [64, 64]
    const float* bias = (const float*)d_in[2];   // [64]
    const int*   ei   = (const int*)d_in[3];     // [2, E] flat

    const int n = in_sizes[0] / D_FEAT;
    const int e = in_sizes[3] / 2;
    const int* row = ei;        // source j
    const int* col = ei + e;    // target i

    // Workspace partition: dinv [n] | hA [n*64] | hB [n*64]
    float* dinv = (float*)d_ws;
    float* hA   = dinv + n;
    float* hB   = hA + (size_t)n * D_FEAT;

    const int B = 256;
    sg_init_deg <<<(n + B - 1) / B, B, 0, stream>>>(dinv, n);
    sg_accum_deg<<<(e + B - 1) / B, B, 0, stream>>>(col, dinv, e);
    sg_rsqrt    <<<(n + B - 1) / B, B, 0, stream>>>(dinv, n);

    const long long nw = (long long)n * D_FEAT;
    const long long ew = (long long)e * D_FEAT;

    // hop 1: hA = self-term(x) + edge scatter(x)
    sg_self <<<(int)((nw + B - 1) / B), B, 0, stream>>>(x, dinv, hA, n);
    sg_edges<<<(int)((ew + B - 1) / B), B, 0, stream>>>(row, col, dinv, x, hA, e);
    // hop 2: hB = self-term(hA) + edge scatter(hA)
    sg_self <<<(int)((nw + B - 1) / B), B, 0, stream>>>(hA, dinv, hB, n);
    sg_edges<<<(int)((ew + B - 1) / B), B, 0, stream>>>(row, col, dinv, hA, hB, e);

    // out = relu(hB @ W^T + b)
    const int mt = n / 16;
    if (mt > 0)
        sg_gemm_relu_wmma<<<mt, 128, 0, stream>>>(hB, W, bias, (float*)d_out);
    const int tail = n - mt * 16;
    if (tail > 0)
        sg_gemm_relu_tail<<<tail, D_FEAT, 0, stream>>>(hB, W, bias, (float*)d_out, mt * 16, n);
}